// _CorrelationNet_86715389706815
// MI455X (gfx1250) — compile-verified
//
#include <hip/hip_runtime.h>

// corr[b,t,w] = dot(x1[b,t,:], x0[b,(t+w-5) mod CT,:]) / C
// B=8, CT=4096, C=64, W=11.  Memory-bound (18.2 MB @ 23.3 TB/s ~ 0.8us),
// so we keep exact f32 math via V_WMMA_F32_16X16X4_F32 and read each input
// byte exactly once.  Per 16-row diagonal block the band is exactly covered
// by two 16x16 WMMA output tiles: X1 * X0[t0-8:t0+8]^T and X1 * X0[t0+8:t0+24]^T.

typedef float v2f __attribute__((ext_vector_type(2)));
typedef float v8f __attribute__((ext_vector_type(8)));

#define BATCH 8
#define CT    4096
#define CHAN  64
#define WIN   11

__global__ __launch_bounds__(256)
void corr_wmma_f32_kernel(const float* __restrict__ x0,
                          const float* __restrict__ x1,
                          float* __restrict__ out) {
  // one wave (32 lanes) handles one 16-row block of t
  const int wave = blockIdx.x * 8 + (threadIdx.x >> 5);
  const int lane = threadIdx.x & 31;
  const int li   = lane & 15;   // row-in-tile for A, col-in-tile for B
  const int half = lane >> 4;   // selects K-pair {0,1} vs {2,3}

  const int b  = wave >> 8;             // CT/16 = 256 blocks per batch
  const int t0 = (wave & 255) << 4;

  // source rows (cyclic): A from x1, B-left/right tiles from x0
  const int rowA = t0 + li;
  const int rowL = (t0 - 8 + li) & (CT - 1);
  const int rowR = (t0 + 8 + li) & (CT - 1);

  const float* pA = x1 + ((size_t)b * CT + rowA) * CHAN + half * 2;
  const float* pL = x0 + ((size_t)b * CT + rowL) * CHAN + half * 2;
  const float* pR = x0 + ((size_t)b * CT + rowR) * CHAN + half * 2;

  v8f accL = {};
  v8f accR = {};

  // C = 64 channels = 16 K-steps of the f32 16x16x4 WMMA
#pragma unroll
  for (int k = 0; k < 16; ++k) {
    const v2f a  = *(const v2f*)(pA + 4 * k);   // A fragment: x1 rows
    const v2f bl = *(const v2f*)(pL + 4 * k);   // B fragment: x0 rows (transposed product)
    const v2f br = *(const v2f*)(pR + 4 * k);
    accL = __builtin_amdgcn_wmma_f32_16x16x4_f32(
        false, a, false, bl, (short)0, accL, false, false);
    accR = __builtin_amdgcn_wmma_f32_16x16x4_f32(
        false, a, false, br, (short)0, accR, false, false);
  }

  // D layout (f32 16x16): lanes 0-15 -> M = v, lanes 16-31 -> M = v + 8; N = li
  const float scale = 1.0f / (float)CHAN;
  const int   n     = li;
  float* obase = out + ((size_t)b * CT + t0) * WIN;
#pragma unroll
  for (int v = 0; v < 8; ++v) {
    const int m = v + half * 8;
    // left tile: j = i + w + 3  -> covers i+w <= 12
    int w = n - m - 3;
    if (0 <= w && w <= 10) obase[(size_t)m * WIN + w] = accL[v] * scale;
    // right tile: j = i + w - 13 -> covers i+w >= 13
    w = n - m + 13;
    if (0 <= w && w <= 10) obase[(size_t)m * WIN + w] = accR[v] * scale;
  }
}

extern "C" void kernel_launch(void* const* d_in, const int* in_sizes, int n_in,
                              void* d_out, int out_size, void* d_ws, size_t ws_size,
                              hipStream_t stream) {
  (void)in_sizes; (void)n_in; (void)out_size; (void)d_ws; (void)ws_size;
  const float* x0 = (const float*)d_in[0];
  const float* x1 = (const float*)d_in[1];
  float* out = (float*)d_out;

  const int waves_total = BATCH * (CT / 16);  // 2048 waves, one per 16-row block
  dim3 block(256);                            // 8 wave32 per block
  dim3 grid(waves_total / 8);                 // 256 blocks
  corr_wmma_f32_kernel<<<grid, block, 0, stream>>>(x0, x1, out);
}